// NoisyTopExpertsPerItemRouter_90048284328016
// MI455X (gfx1250) — compile-verified
//
#include <hip/hip_runtime.h>
#include <hip/hip_bf16.h>

typedef __attribute__((ext_vector_type(2))) float v2f;
typedef __attribute__((ext_vector_type(8))) float v8f;
typedef int vsi4 __attribute__((vector_size(16)));   // matches builtin param type

#define AS1 __attribute__((address_space(1)))
#define AS3 __attribute__((address_space(3)))

#define G_ 64
#define S_ 1024
#define D_ 768
#define E_ 64
#define B_ 32
#define P_ 512
#define M_ (G_*S_)            // 65536 rows

// ---- workspace layout (float offsets) ----
#define WS_LOGITS ((size_t)0)
#define WS_GSM    ((size_t)M_*E_)            // 4194304
#define WS_BJSD   ((size_t)2*(size_t)M_*E_)  // 8388608
#define WS_BCNT   (WS_BJSD + 32)
#define WS_PCC    (WS_BCNT + 32)

// ---- output layout (float offsets) ----
#define OUT_COMBINE 0
#define OUT_IDX     (M_*2)        // 131072
#define OUT_SUM     (M_*4)        // 262144
#define OUT_IMP     (OUT_SUM + G_)
#define OUT_LOAD    (OUT_IMP + G_)
#define OUT_GSH     (OUT_LOAD + G_)
#define OUT_JSD     (OUT_GSH + G_)

#if __has_builtin(__builtin_amdgcn_global_load_async_to_lds_b128) && \
    __has_builtin(__builtin_amdgcn_global_load_async_to_lds_b32)
#define ASYNC_LDS 1
#else
#define ASYNC_LDS 0
#endif

__device__ __forceinline__ void wait_async_lds() {
#if ASYNC_LDS
#if __has_builtin(__builtin_amdgcn_s_wait_asynccnt)
    __builtin_amdgcn_s_wait_asynccnt(0);
#else
    asm volatile("s_wait_asynccnt 0x0" ::: "memory");
#endif
#endif
}

// ---------------- helpers ----------------
__device__ __forceinline__ float wsum(float v) {
#pragma unroll
    for (int o = 16; o >= 1; o >>= 1) v += __shfl_xor(v, o, 32);
    return v;
}
__device__ __forceinline__ float wmax(float v) {
#pragma unroll
    for (int o = 16; o >= 1; o >>= 1) v = fmaxf(v, __shfl_xor(v, o, 32));
    return v;
}
// merge other lane's (w1,j1,w2,j2) top-2 into mine; ties -> lower index (jax top_k)
__device__ __forceinline__ void top2_merge(float& v1, int& i1, float& v2, int& i2,
                                           float w1, int j1, float w2, int j2) {
    bool otherTop = (w1 > v1) || (w1 == v1 && j1 < i1);
    if (otherTop) {
        float nv2; int ni2;
        bool mineBeats = (v1 > w2) || (v1 == w2 && i1 < j2);
        if (mineBeats) { nv2 = v1; ni2 = i1; } else { nv2 = w2; ni2 = j2; }
        v1 = w1; i1 = j1; v2 = nv2; i2 = ni2;
    } else {
        bool other2 = (w1 > v2) || (w1 == v2 && j1 < i2);
        if (other2) { v2 = w1; i2 = j1; }
    }
}

// ---------------- kernel 0: init small accumulators ----------------
__global__ void k_init(float* __restrict__ ws) {
    int t = threadIdx.x;
    if (t < 32) ws[WS_BJSD + t] = 0.0f;
    if (t < 32) ws[WS_BCNT + t] = 0.0f;
    if (t == 0) ws[WS_PCC] = 0.0f;
}

// ---------------- kernel 1: WMMA fp32 GEMM  logits = inputs @ W ----------------
// grid: 512 blocks (128 rows each), 256 threads = 8 waves; wave -> 16 rows x 64 cols
// Double-buffered LDS panels; async global->LDS staging overlapped with WMMA.

// stage one K-panel (32 wide) of A (128x32) and W^T (64x32) into LDS buffers
__device__ __forceinline__ void stage_panels(const float* __restrict__ A,
                                             const float* __restrict__ Wg,
                                             int rowBlock, int kc, int tid,
                                             float (*As)[36], float (*Ws)[34]) {
#pragma unroll
    for (int i = 0; i < 4; ++i) {
        int seg = tid + 256 * i;            // 0..1023
        int r   = seg >> 3;
        int c4  = (seg & 7) << 2;
        const float* src = A + (size_t)(rowBlock + r) * D_ + (kc + c4);
#if ASYNC_LDS
        __builtin_amdgcn_global_load_async_to_lds_b128(
            (AS1 vsi4*)src, (AS3 vsi4*)&As[r][c4], 0, 0);
#else
        float4 v = *(const float4*)src;
        As[r][c4 + 0] = v.x; As[r][c4 + 1] = v.y;
        As[r][c4 + 2] = v.z; As[r][c4 + 3] = v.w;
#endif
    }
#pragma unroll
    for (int i = 0; i < 8; ++i) {
        int idx = tid + 256 * i;            // 0..2047
        int kk  = idx >> 6;
        int n   = idx & 63;
        const float* src = Wg + (size_t)(kc + kk) * E_ + n;
#if ASYNC_LDS
        __builtin_amdgcn_global_load_async_to_lds_b32(
            (AS1 int*)src, (AS3 int*)&Ws[n][kk], 0, 0);
#else
        Ws[n][kk] = *src;
#endif
    }
}

__device__ __forceinline__ void gemm_panel(const float (*As)[36], const float (*Ws)[34],
                                           int wave, int lane, v8f acc[4]) {
    const int half = lane >> 4;
    const int l16  = lane & 15;
#pragma unroll
    for (int kk = 0; kk < 32; kk += 4) {
        const int ak = kk + 2 * half;       // lane half -> K=0,1 vs K=2,3 (ISA A 16x4)
        float2 ta = *(const float2*)&As[(wave << 4) + l16][ak];
        v2f a; a[0] = ta.x; a[1] = ta.y;
#pragma unroll
        for (int t = 0; t < 4; ++t) {       // 4 N-tiles of 16
            float2 tb = *(const float2*)&Ws[t * 16 + l16][ak];
            v2f b; b[0] = tb.x; b[1] = tb.y;
            acc[t] = __builtin_amdgcn_wmma_f32_16x16x4_f32(
                false, a, false, b, (short)0, acc[t], false, false);
        }
    }
}

__global__ __launch_bounds__(256) void k_gemm(const float* __restrict__ A,
                                              const float* __restrict__ Wg,
                                              float* __restrict__ logits) {
    __shared__ float As[2][128][36];   // stride 36: rows 16B aligned (b128), banks 4*(9r%16)
    __shared__ float Ws[2][64][34];    // W^T chunk [n][k]; b64-aligned fragment reads

    const int tid  = threadIdx.x;
    const int wave = tid >> 5;
    const int lane = tid & 31;
    const int rowBlock = blockIdx.x * 128;

    v8f acc[4] = {{}, {}, {}, {}};

    stage_panels(A, Wg, rowBlock, 0, tid, As[0], Ws[0]);
    wait_async_lds();
    __syncthreads();

    for (int kc = 0; kc < D_; kc += 32) {
        const int cur = (kc >> 5) & 1;
        const int nxt = cur ^ 1;
        if (kc + 32 < D_)
            stage_panels(A, Wg, rowBlock, kc + 32, tid, As[nxt], Ws[nxt]);  // overlap
        gemm_panel(As[cur], Ws[cur], wave, lane, acc);                       // 32 wmma
        if (kc + 32 < D_) wait_async_lds();
        __syncthreads();
    }

    // store C: VGPR v -> row (v + 8*half), col = t*16 + l16
    const int half = lane >> 4;
    const int l16  = lane & 15;
    const int rowBase = rowBlock + (wave << 4) + (half << 3);
#pragma unroll
    for (int t = 0; t < 4; ++t)
#pragma unroll
        for (int v = 0; v < 8; ++v)
            logits[(size_t)(rowBase + v) * E_ + t * 16 + l16] = acc[t][v];
}

// ---------------- kernel 2: per-group router (softmax, top-2, losses) ----------------
// grid: 64 blocks (one per g), 256 threads = 8 waves; wave handles 128 rows; 2 experts/lane
__global__ __launch_bounds__(256) void k_router(const float* __restrict__ logits,
                                                const float* __restrict__ noise,
                                                float* __restrict__ gsm,
                                                float* __restrict__ out) {
    const int g    = blockIdx.x;
    const int wave = threadIdx.x >> 5;
    const int lane = threadIdx.x & 31;
    const float noise_std = 1.0f / 64.0f;
    const float inv_ns    = 64.0f;
    const float inv_sqrt2 = 0.70710678118654752f;

    __shared__ float simp[64], sp[64], sgn[64], scnt[64];
    if (threadIdx.x < 64) { simp[threadIdx.x] = 0.f; sp[threadIdx.x] = 0.f;
                            sgn[threadIdx.x] = 0.f; scnt[threadIdx.x] = 0.f; }
    __syncthreads();

    float imp0 = 0.f, imp1 = 0.f, p0s = 0.f, p1s = 0.f, gn0 = 0.f, gn1 = 0.f;
    float c0 = 0.f, c1 = 0.f;
    const int e0 = 2 * lane, e1 = 2 * lane + 1;

    for (int j = wave; j < S_; j += 8) {
        const size_t row = (size_t)g * S_ + j;
        float2 l  = *(const float2*)(logits + row * E_ + e0);
        float2 nz = *(const float2*)(noise  + row * E_ + e0);
        float nl0 = l.x + noise_std * nz.x;
        float nl1 = l.y + noise_std * nz.y;

        // clean softmax
        float m  = wmax(fmaxf(l.x, l.y));
        float x0 = __expf(l.x - m), x1 = __expf(l.y - m);
        float inv = 1.0f / wsum(x0 + x1);
        float g0 = x0 * inv, g1 = x1 * inv;
        *(float2*)(gsm + row * E_ + e0) = make_float2(g0, g1);
        imp0 += g0; imp1 += g1;

        // noisy softmax
        float mn  = wmax(fmaxf(nl0, nl1));
        float y0  = __expf(nl0 - mn), y1 = __expf(nl1 - mn);
        float invn = 1.0f / wsum(y0 + y1);
        gn0 += y0 * invn; gn1 += y1 * invn;

        // top-2 of noisy logits (softmax is monotonic -> same order as noisy softmax)
        float v1, v2; int i1, i2;
        if ((nl0 > nl1) || (nl0 == nl1)) { v1 = nl0; i1 = e0; v2 = nl1; i2 = e1; }
        else                             { v1 = nl1; i1 = e1; v2 = nl0; i2 = e0; }
#pragma unroll
        for (int o = 16; o >= 1; o >>= 1) {
            float w1 = __shfl_xor(v1, o, 32); int j1 = __shfl_xor(i1, o, 32);
            float w2 = __shfl_xor(v2, o, 32); int j2 = __shfl_xor(i2, o, 32);
            top2_merge(v1, i1, v2, i2, w1, j1, w2, j2);
        }
        // combine weights = noisy softmax at top-2
        if (lane == 0) {
            float cw1 = __expf(v1 - mn) * invn;
            float cw2 = __expf(v2 - mn) * invn;
            out[OUT_COMBINE + row * 2 + 0] = cw1;
            out[OUT_COMBINE + row * 2 + 1] = cw2;
            out[OUT_IDX     + row * 2 + 0] = (float)i1;
            out[OUT_IDX     + row * 2 + 1] = (float)i2;
        }
        if (i1 == e0) c0 += 1.f; else if (i1 == e1) c1 += 1.f;

        // load-loss: p = 1 - Phi((thr - logit)/noise_std), thr = 2nd noisy logit
        p0s += 0.5f * erfcf((v2 - l.x) * inv_ns * inv_sqrt2);
        p1s += 0.5f * erfcf((v2 - l.y) * inv_ns * inv_sqrt2);
    }

    atomicAdd(&simp[e0], imp0); atomicAdd(&simp[e1], imp1);
    atomicAdd(&sp[e0],   p0s);  atomicAdd(&sp[e1],   p1s);
    atomicAdd(&sgn[e0],  gn0);  atomicAdd(&sgn[e1],  gn1);
    atomicAdd(&scnt[e0], c0);   atomicAdd(&scnt[e1], c1);
    __syncthreads();

    if (wave == 0) {
        // importance: var/mean^2 over 64 expert sums
        float a0 = simp[e0], a1 = simp[e1];
        float s1 = wsum(a0 + a1), s2 = wsum(a0 * a0 + a1 * a1);
        float mean = s1 / 64.f, var = s2 / 64.f - mean * mean;
        float imp_loss = var / (mean * mean);

        float q0 = sp[e0] / (float)S_, q1 = sp[e1] / (float)S_;
        float t1 = wsum(q0 + q1), t2 = wsum(q0 * q0 + q1 * q1);
        float pm = t1 / 64.f, pv = t2 / 64.f - pm * pm;
        float load_loss = pv / (pm * pm);

        // gshard: mean(mean_top1 * mean_gates) * E^2
        float gsum = wsum(scnt[e0] * sgn[e0] + scnt[e1] * sgn[e1]);
        float gshard = gsum * (64.f / ((float)S_ * (float)S_));

        if (lane == 0) {
            out[OUT_IMP  + g] = imp_loss;
            out[OUT_LOAD + g] = load_loss;
            out[OUT_GSH  + g] = gshard;
        }
    }
}

// ---------------- kernel 3: JSD over patch correspondences ----------------
// one wave per pair; 16384 pairs -> 2048 blocks x 8 waves
__global__ __launch_bounds__(256) void k_jsd(const float* __restrict__ gsm,
                                             const int* __restrict__ pc,
                                             float* __restrict__ ws) {
    const int wave = threadIdx.x >> 5;
    const int lane = threadIdx.x & 31;
    const int pair = blockIdx.x * 8 + wave;      // 0..16383
    const int b = pair >> 9;                     // /512
    const int p = pair & 511;

    const int i0 = pc[((size_t)b * P_ + p) * 2 + 0];
    const int i1 = pc[((size_t)b * P_ + p) * 2 + 1];

    if (lane == 0) {
        float cnt = (i0 > -1 ? 0.5f : 0.f) + (i1 > -1 ? 0.5f : 0.f);
        if (cnt > 0.f) atomicAdd(&ws[WS_PCC], cnt);
        if (i0 > -1)   atomicAdd(&ws[WS_BCNT + b], 1.0f);
    }

    const int e0 = 2 * lane;
    float2 pg = make_float2(0.f, 0.f), qg = make_float2(0.f, 0.f);
    if (i0 >= 0 && i0 < S_)
        pg = *(const float2*)(gsm + ((size_t)(b * 2 + 0) * S_ + i0) * E_ + e0);
    if (i1 >= 0 && i1 < S_)
        qg = *(const float2*)(gsm + ((size_t)(b * 2 + 1) * S_ + i1) * E_ + e0);

    float m0 = 0.5f * (pg.x + qg.x), m1 = 0.5f * (pg.y + qg.y);
    float kp = pg.x * __logf(pg.x / (m0 + 1e-7f) + 1e-7f)
             + pg.y * __logf(pg.y / (m1 + 1e-7f) + 1e-7f);
    float kq = qg.x * __logf(qg.x / (m0 + 1e-7f) + 1e-7f)
             + qg.y * __logf(qg.y / (m1 + 1e-7f) + 1e-7f);
    float tot = wsum(kp + kq);
    if (lane == 0) atomicAdd(&ws[WS_BJSD + b], 0.5f * tot);
}

// ---------------- kernel 4: finalize jsd scalar + sum_loss ----------------
__global__ void k_final(const float* __restrict__ ws, float* __restrict__ out) {
    __shared__ float sjsd;
    const int t = threadIdx.x;
    if (t < 32) {
        float v = ws[WS_BJSD + t] * ws[WS_BCNT + t];   // val_b * count(i0>-1)
        v = wsum(v);
        if (t == 0) {
            float pcc = ws[WS_PCC];
            float jsd = (pcc > 0.f) ? (v / pcc) : 0.f;
            out[OUT_JSD] = jsd;
            sjsd = jsd;
        }
    }
    __syncthreads();
    if (t < 64)
        out[OUT_SUM + t] = out[OUT_IMP + t] + out[OUT_LOAD + t] + sjsd; // GSHARD_W = 0
}

extern "C" void kernel_launch(void* const* d_in, const int* in_sizes, int n_in,
                              void* d_out, int out_size, void* d_ws, size_t ws_size,
                              hipStream_t stream) {
    (void)in_sizes; (void)n_in; (void)out_size; (void)ws_size;
    const float* inputs = (const float*)d_in[0];   // [64,1024,768]
    const float* W      = (const float*)d_in[1];   // [768,64]
    const float* noise  = (const float*)d_in[2];   // [64,1024,64]
    const int*   pc     = (const int*)d_in[3];     // [32,512,2]
    float* out = (float*)d_out;
    float* ws  = (float*)d_ws;

    float* logits = ws + WS_LOGITS;
    float* gsm    = ws + WS_GSM;

    k_init<<<1, 64, 0, stream>>>(ws);
    k_gemm<<<M_ / 128, 256, 0, stream>>>(inputs, W, logits);
    k_router<<<G_, 256, 0, stream>>>(logits, noise, gsm, out);
    k_jsd<<<(B_ * P_) / 8, 256, 0, stream>>>(gsm, pc, ws);
    k_final<<<1, 64, 0, stream>>>(ws, out);
}